// RoIAlignAvg_3d_18073222382195
// MI455X (gfx1250) — compile-verified
//
#include <hip/hip_runtime.h>

// RoIAlign-3D + fused avg_pool3d(k2,s1) for MI455X (gfx1250, wave32).
// Separable formulation: per-roi pooled+masked weight matrices WT(4x9),
// WH(7x24), WW(7x24); contraction done with V_WMMA_F32_16X16X4_F32.

typedef __attribute__((ext_vector_type(2))) float v2f;
typedef __attribute__((ext_vector_type(8))) float v8f;

namespace {
constexpr int C = 256, T = 16, H = 64, W = 64;
constexpr int PT = 5, PH = 8, PW = 8;        // roi_align sample counts
constexpr int OT = 4, OH = 7, OW = 7;        // pooled output dims
constexpr int WIN = 24;                      // x/y index window (covers span)
constexpr int TWIN = 9;                      // t index window
constexpr float SSCALE = 1.0f / 16.0f;
constexpr float TSCALE = 1.0f;
constexpr int NWAVES = 4;
constexpr int S1_SZ  = 16 * OW * WIN;        // [c16][wp7][yy24] = 2688 floats
constexpr int ACC_SZ = 16 * OT * OH * OW;    // [c16][tp4][hp7][wp7] = 3136
}

// Build pooled (k2,s1, x0.5) + masked interpolation weight rows for one axis.
// wrows[(n_out-1) x ncols], indices relative to `origin` (window start).
__device__ inline void build_axis(float start, float end, int size, int n_out,
                                  int origin, float* wrows, int ncols) {
  float length = fmaxf(end - start + 1.0f, 1.0f);
  float step = length / (float)(n_out - 1);
  for (int i = 0; i < n_out; ++i) {
    float coord = start + step * (float)i;
    float valid = (coord >= 0.0f && coord < (float)size) ? 1.0f : 0.0f;
    float lof = fminf(fmaxf(floorf(coord), 0.0f), (float)(size - 1));
    int lo = (int)lof;
    int hi = min(lo + 1, size - 1);
    float frac = coord - lof;
    float wlo = (1.0f - frac) * valid;
    float whi = frac * valid;
    int cl = lo - origin, ch = hi - origin;
    if (cl >= 0 && cl < ncols) {
      if (i >= 1)         wrows[(i - 1) * ncols + cl] += 0.5f * wlo;
      if (i <= n_out - 2) wrows[i * ncols + cl]       += 0.5f * wlo;
    }
    if (ch >= 0 && ch < ncols) {
      if (i >= 1)         wrows[(i - 1) * ncols + ch] += 0.5f * whi;
      if (i <= n_out - 2) wrows[i * ncols + ch]       += 0.5f * whi;
    }
  }
}

__global__ __launch_bounds__(128, 1)
void roi_align_pool_wmma(const float* __restrict__ feat,
                         const float* __restrict__ rois,
                         float* __restrict__ out) {
  __shared__ __align__(16) float sWW[OW * WIN];
  __shared__ __align__(16) float sWH[OW * WIN];
  __shared__ __align__(16) float sWT[OT * TWIN];
  __shared__ int sInfo[4];
  __shared__ __align__(16) float sS1[NWAVES * S1_SZ];
  __shared__ __align__(16) float sAcc[NWAVES * ACC_SZ];

  const int r = blockIdx.x;
  const int tid = threadIdx.x;

  for (int i = tid; i < OW * WIN; i += blockDim.x) { sWW[i] = 0.0f; sWH[i] = 0.0f; }
  for (int i = tid; i < OT * TWIN; i += blockDim.x) sWT[i] = 0.0f;
  __syncthreads();

  if (tid == 0) {
    const float* rp = rois + (size_t)r * 7;
    int b    = (int)rp[0];
    float x1 = rp[1] * SSCALE, y1 = rp[2] * SSCALE;
    float x2 = rp[3] * SSCALE, y2 = rp[4] * SSCALE;
    float t1 = rp[5] * TSCALE, t2 = rp[6] * TSCALE;
    // Window origins: all used lo/hi indices land in [origin, origin+ncols-1],
    // and the whole window is in-bounds (so streamed loads never fault).
    int xx0 = max(0, min((int)floorf(x1), W - WIN));
    int yy0 = max(0, min((int)floorf(y1), H - WIN));
    int tt0 = max(0, min((int)floorf(t1), T - TWIN));
    sInfo[0] = b; sInfo[1] = xx0; sInfo[2] = yy0; sInfo[3] = tt0;
    build_axis(x1, x2, W, PW, xx0, sWW, WIN);
    build_axis(y1, y2, H, PH, yy0, sWH, WIN);
    build_axis(t1, t2, T, PT, tt0, sWT, TWIN);
  }
  __syncthreads();

  const int b   = sInfo[0];
  const int xx0 = sInfo[1];
  const int yy0 = sInfo[2];
  const int tt0 = sInfo[3];

  const int wid  = tid >> 5;    // wave id (wave32)
  const int lane = tid & 31;
  const int n    = lane & 15;   // N column (B/C/D) or M row (A)
  const int half = lane >> 4;   // K half select
  const int wpc  = min(n, OW - 1); // clamp for B reads (cols >=7 are dont-care)

  // B operands (ISA 4x16 layout: lane<16 -> {B[k0][n],B[k1][n]}, lane>=16 -> {B[k2][n],B[k3][n]})
  v2f Bx[6], By[6];
#pragma unroll
  for (int j = 0; j < 6; ++j) {
    int k = 4 * j + 2 * half;
    v2f bx; bx.x = sWW[wpc * WIN + k]; bx.y = sWW[wpc * WIN + k + 1];
    v2f by; by.x = sWH[wpc * WIN + k]; by.y = sWH[wpc * WIN + k + 1];
    Bx[j] = bx; By[j] = by;
  }

  float* s1  = sS1  + wid * S1_SZ;
  float* acc = sAcc + wid * ACC_SZ;

  for (int cb = wid; cb < 16; cb += NWAVES) {
    const int c0 = cb * 16;
    for (int i = lane; i < ACC_SZ; i += 32) acc[i] = 0.0f;

    for (int tj = 0; tj < TWIN; ++tj) {
      const int tt = tt0 + tj;
      // ---- X stage: D[c,wp] = sum_xx f[c,tt,yy,xx0+k] * WW^T[k,wp] ----
      const float* fbase = feat
          + ((((size_t)b * C + (size_t)(c0 + n)) * T + (size_t)tt) * (H * W))
          + (size_t)yy0 * W + xx0 + 2 * half;
      for (int yy = 0; yy < WIN; ++yy) {
        const float* frow = fbase + yy * W;
        v8f d = {0.f, 0.f, 0.f, 0.f, 0.f, 0.f, 0.f, 0.f};
#pragma unroll
        for (int j = 0; j < 6; ++j) {
          v2f a; a.x = frow[4 * j]; a.y = frow[4 * j + 1];
          d = __builtin_amdgcn_wmma_f32_16x16x4_f32(
              false, a, false, Bx[j], (short)0, d, false, false);
        }
        if (n < OW) {  // store valid D columns: s1[c][wp=n][yy]
          float* sp = s1 + (half * 8) * (OW * WIN) + n * WIN + yy;
#pragma unroll
          for (int v = 0; v < 8; ++v) sp[v * (OW * WIN)] = d[v];
        }
      }

      // ---- Y stage + T fold ----
      const float wtj0 = sWT[0 * TWIN + tj];
      const float wtj1 = sWT[1 * TWIN + tj];
      const float wtj2 = sWT[2 * TWIN + tj];
      const float wtj3 = sWT[3 * TWIN + tj];
      for (int wp = 0; wp < OW; ++wp) {
        v8f d = {0.f, 0.f, 0.f, 0.f, 0.f, 0.f, 0.f, 0.f};
#pragma unroll
        for (int j = 0; j < 6; ++j) {
          int k = 4 * j + 2 * half;
          const float* p = s1 + n * (OW * WIN) + wp * WIN + k; // A row M=c=n
          v2f a; a.x = p[0]; a.y = p[1];
          d = __builtin_amdgcn_wmma_f32_16x16x4_f32(
              false, a, false, By[j], (short)0, d, false, false);
        }
        if (n < OH) {  // D col = hp = n; rows = channels
#pragma unroll
          for (int v = 0; v < 8; ++v) {
            float dv = d[v];
            int idx = (half * 8 + v) * (OT * OH * OW) + n * OW + wp;
            if (wtj0 != 0.0f) acc[idx + 0 * (OH * OW)] += wtj0 * dv;
            if (wtj1 != 0.0f) acc[idx + 1 * (OH * OW)] += wtj1 * dv;
            if (wtj2 != 0.0f) acc[idx + 2 * (OH * OW)] += wtj2 * dv;
            if (wtj3 != 0.0f) acc[idx + 3 * (OH * OW)] += wtj3 * dv;
          }
        }
      }
    }

    // acc layout [c][tp][hp][wp] matches output layout exactly -> flat copy.
    float* op = out + (size_t)r * (C * OT * OH * OW) + (size_t)c0 * (OT * OH * OW);
    for (int i = lane; i < ACC_SZ; i += 32) op[i] = acc[i];
  }
}

extern "C" void kernel_launch(void* const* d_in, const int* in_sizes, int n_in,
                              void* d_out, int out_size, void* d_ws, size_t ws_size,
                              hipStream_t stream) {
  const float* feat = (const float*)d_in[0];   // (4,256,16,64,64) f32
  const float* rois = (const float*)d_in[1];   // (N,7) f32
  float* out = (float*)d_out;                  // (N,256,4,7,7) f32
  const int nrois = in_sizes[1] / 7;
  roi_align_pool_wmma<<<nrois, 128, 0, stream>>>(feat, rois, out);
}